// HistByProfMultiChannel_42425686950255
// MI455X (gfx1250) — compile-verified
//
#include <hip/hip_runtime.h>
#include <hip/hip_bf16.h>
#include <stdint.h>

// HistByProfMultiChannel for MI455X (gfx1250, wave32).
//  - Async global->LDS staging of the per-(bt,c) pixel chunk (ASYNCcnt path)
//  - Trans-unit-native math: exp(-0.5 z^2) = exp2(-0.72134752 * z^2) -> v_exp_f32
//  - Per-lane register accumulation (compute-bound on trans unit, ~70M exps total)
//  - Final 256-thread x 11-bin reduction done as ones(16x4) x P(4x16) chained
//    V_WMMA_F32_16X16X4_F32 on wave 0 (A==ones makes the K-slot mapping immaterial)

typedef float v2f __attribute__((ext_vector_type(2)));
typedef float v8f __attribute__((ext_vector_type(8)));

#define HW      3136      // 56*56 pixels per (bt,c)
#define NE      10        // number of edges
#define NBINS   11        // NE + 1
#define PSTR    16        // padded LDS row stride for partials
#define NTHREADS 256

__global__ __launch_bounds__(NTHREADS)
void hist_by_prof_kernel(const float* __restrict__ x,
                         const float* __restrict__ edges,
                         float* __restrict__ out)
{
    __shared__ __align__(16) float sx[HW];           // staged pixels (12544 B)
    __shared__ __align__(16) float P[NTHREADS * PSTR]; // per-thread partials (16 KB)

    const int tid = threadIdx.x;
    const int blk = blockIdx.x;        // bt*128 + c, x chunk is contiguous
    const int c   = blk & 127;

    // ---------------- Phase 1: async HBM -> LDS copy (CDNA5 async path) ------
    const float* src = x + (size_t)blk * HW;
    for (int i = tid; i < HW / 4; i += NTHREADS) {
        uint32_t lds_addr = (uint32_t)(uintptr_t)(&sx[i * 4]); // wave-relative LDS offset
        uint32_t byte_off = (uint32_t)(i * 16);
        asm volatile("global_load_async_to_lds_b128 %0, %1, %2"
                     :: "v"(lds_addr), "v"(byte_off), "s"(src)
                     : "memory");
    }
    asm volatile("s_wait_asynccnt 0" ::: "memory");
    __syncthreads();

    // ---------------- Per-channel bin parameters (uniform across block) ------
    const float* e = edges + c * NE;
    float ev[NE];
    #pragma unroll
    for (int i = 0; i < NE; ++i) ev[i] = e[i];

    float mu[NE], isg[NE];
    {
        float s0 = (ev[0] - ev[1]) * (1.0f / 3.0f) + 1e-6f;  // bin 0: mu=e0
        mu[0]  = ev[0];
        isg[0] = __builtin_amdgcn_rcpf(s0);
        #pragma unroll
        for (int i = 1; i < NE; ++i) {
            mu[i] = 0.5f * (ev[i - 1] + ev[i]);
            float s = (ev[i - 1] - ev[i]) * (1.0f / 3.0f) + 1e-6f;
            isg[i] = __builtin_amdgcn_rcpf(s);
        }
    }
    const float e_last = ev[NE - 1];

    // ---------------- Phase 2: per-lane accumulation over own pixels ---------
    float acc[NBINS];
    #pragma unroll
    for (int i = 0; i < NBINS; ++i) acc[i] = 0.0f;

    const float kGauss = -0.72134752044448169f;   // -0.5 * log2(e)
    const float kSigm  =  28.853900817779268f;    //  20  * log2(e)

    for (int p = tid; p < HW; p += NTHREADS) {
        float xv = sx[p];                          // ds_load_b32, conflict-free
        #pragma unroll
        for (int i = 0; i < NE; ++i) {
            float t = (xv - mu[i]) * isg[i];
            acc[i] += __builtin_amdgcn_exp2f(t * t * kGauss);   // v_exp_f32
        }
        // sigmoid tail: 1/(1+exp(-20*(x-e_last)))
        acc[NE] += __builtin_amdgcn_rcpf(
            1.0f + __builtin_amdgcn_exp2f((e_last - xv) * kSigm));
    }

    // ---------------- Phase 3: partials -> LDS matrix ------------------------
    #pragma unroll
    for (int i = 0; i < NBINS; ++i) P[tid * PSTR + i] = acc[i];
    #pragma unroll
    for (int i = NBINS; i < PSTR; ++i) P[tid * PSTR + i] = 0.0f;  // branch-free B reads
    __syncthreads();

    // ---------------- Phase 4: wave0 WMMA reduction: ones(16x4) x P(4x16) ----
    // EXEC is all-ones inside this wave-uniform branch (WMMA requirement).
    if (tid < 32) {
        const int n = tid & 15;     // bin column (N = lane % 16)
        const int h = tid >> 4;     // half-wave selects K rows
        v2f ones; ones.x = 1.0f; ones.y = 1.0f;
        v8f d = {};                 // zero C accumulator
        for (int s = 0; s < NTHREADS; s += 4) {
            int r0 = s + 2 * h;     // any (h,j)->K bijection is valid since A==1
            v2f b;
            b.x = P[(r0 + 0) * PSTR + n];
            b.y = P[(r0 + 1) * PSTR + n];
            // D = ones * B + C  => every row of D = per-bin sum over 4 threads
            d = __builtin_amdgcn_wmma_f32_16x16x4_f32(
                    /*neg_a=*/false, ones, /*neg_b=*/false, b,
                    /*c_mod=*/(short)0, d, /*reuse_a=*/false, /*reuse_b=*/false);
        }
        // C/D layout: lane l, VGPR0 holds row (l<16 ? 0 : 8), col l%16.
        // All rows identical -> lanes 0..10 hold the 11 bin totals in d[0].
        if (tid < NBINS) out[(size_t)blk * NBINS + tid] = d[0];
    }
}

extern "C" void kernel_launch(void* const* d_in, const int* in_sizes, int n_in,
                              void* d_out, int out_size, void* d_ws, size_t ws_size,
                              hipStream_t stream) {
    const float* x     = (const float*)d_in[0];   // (16,128,56,56) f32
    const float* edges = (const float*)d_in[1];   // (128,10) f32
    float* out = (float*)d_out;                   // (16,128,11) f32

    const int nblocks = in_sizes[0] / HW;         // 16*128 = 2048
    hipLaunchKernelGGL(hist_by_prof_kernel,
                       dim3(nblocks), dim3(NTHREADS), 0, stream,
                       x, edges, out);
}